// custom_model_38465727103175
// MI455X (gfx1250) — compile-verified
//
#include <hip/hip_runtime.h>
#include <cmath>

#define EPS_BN 1e-5f

typedef __attribute__((ext_vector_type(2))) float v2f;
typedef __attribute__((ext_vector_type(8))) float v8f;

static __device__ __forceinline__ v8f wmma4(v2f a, v2f b, v8f c) {
  // V_WMMA_F32_16X16X4_F32 : D = A(16x4) * B(4x16) + C(16x16), all fp32
  return __builtin_amdgcn_wmma_f32_16x16x4_f32(false, a, false, b, (short)0, c,
                                               false, false);
}

// ---------------------------------------------------------------------------
// Implicit-GEMM convolution: Y[N,Cout,Ho,Wo] = W[Cout,Cin,KH,KW] * im2col(X)
// One wave computes a 16(Cout) x 64(pixels) tile via 4 accumulators.
// Per-k im2col decode cached in LDS (offset + fy/fx), shared by all 8 waves.
// Wave-uniform "interior" fast path removes all per-tap bounds checks for
// tiles whose entire receptive field is in-bounds.
// ---------------------------------------------------------------------------
__global__ __launch_bounds__(256) void conv_wmma(
    const float* __restrict__ X, const float* __restrict__ Wt,
    float* __restrict__ Y, int N, int Cin, int Hin, int Win, int Cout, int KH,
    int KW, int stride, int pad, int Hout, int Wout) {
  __shared__ int sOff[4608];
  __shared__ int sFF[4608];
  const int KHW = KH * KW;
  const int K = Cin * KHW;
  for (int k = threadIdx.x; k < K; k += 256) {
    int ci = k / KHW;
    int r = k - ci * KHW;
    int fy = r / KW;
    int fx = r - fy * KW;
    sOff[k] = (ci * Hin + fy) * Win + fx;
    sFF[k] = (fy << 16) | fx;
  }
  __syncthreads();

  const int lane = threadIdx.x & 31;
  const int wave = threadIdx.x >> 5;
  const int mrow = lane & 15;
  const int khalf = (lane >> 4) << 1;  // 0 or 2
  const long HWo = (long)Hout * Wout;
  const long Ngemm = (long)N * HWo;
  const int coBase = blockIdx.y * 16;
  const float* Arow0 = Wt + (long)(coBase + mrow) * K;
  __builtin_prefetch(Arow0, 0, 3);

  long pixBase = ((long)blockIdx.x * 8 + wave) * 64 + (lane & 15);
  bool valid[4];
  long base[4], outBase[4];
  int h0[4], w0[4];
  bool allInt = true;
#pragma unroll
  for (int sub = 0; sub < 4; ++sub) {
    long pix = pixBase + sub * 16;
    valid[sub] = pix < Ngemm;
    long pc = valid[sub] ? pix : 0;
    int n_img = (int)(pc / HWo);
    int hw = (int)(pc % HWo);
    int ho = hw / Wout;
    int wo = hw - ho * Wout;
    int hh = ho * stride - pad;
    int wwv = wo * stride - pad;
    h0[sub] = valid[sub] ? hh : -1000000;  // force OOB for dummy pixels
    w0[sub] = wwv;
    base[sub] = (long)n_img * Cin * Hin * Win + (long)hh * Win + wwv;
    outBase[sub] = (long)n_img * Cout * HWo + hw;
    allInt = allInt && valid[sub] && hh >= 0 && (hh + KH) <= Hin &&
             wwv >= 0 && (wwv + KW) <= Win;
  }
  const bool fast = __all((int)allInt);
  const int K4 = K & ~3;

  v8f acc[4] = {};
  if (fast) {
    // Interior tile: no bounds checks, no K guards inside the main loop.
    for (int k0 = 0; k0 < K4; k0 += 4) {
      int ka = k0 + khalf;
      v2f a;
      a.x = Arow0[ka];
      a.y = Arow0[ka + 1];
      int o0 = sOff[ka];
      int o1 = sOff[ka + 1];
#pragma unroll
      for (int sub = 0; sub < 4; ++sub) {
        v2f b;
        b.x = X[base[sub] + o0];
        b.y = X[base[sub] + o1];
        acc[sub] = wmma4(a, b, acc[sub]);
      }
    }
  } else {
    // Boundary tile: per-tap zero-padding checks (K guards still hoisted).
    for (int k0 = 0; k0 < K4; k0 += 4) {
      int ka = k0 + khalf;
      v2f a;
      a.x = Arow0[ka];
      a.y = Arow0[ka + 1];
      int o0 = sOff[ka], ff0 = sFF[ka];
      int o1 = sOff[ka + 1], ff1 = sFF[ka + 1];
      int fy0 = ff0 >> 16, fx0 = ff0 & 0xffff;
      int fy1 = ff1 >> 16, fx1 = ff1 & 0xffff;
#pragma unroll
      for (int sub = 0; sub < 4; ++sub) {
        v2f b;
        b.x = 0.f;
        b.y = 0.f;
        int hi0 = h0[sub] + fy0, wi0 = w0[sub] + fx0;
        if ((unsigned)hi0 < (unsigned)Hin && (unsigned)wi0 < (unsigned)Win)
          b.x = X[base[sub] + o0];
        int hi1 = h0[sub] + fy1, wi1 = w0[sub] + fx1;
        if ((unsigned)hi1 < (unsigned)Hin && (unsigned)wi1 < (unsigned)Win)
          b.y = X[base[sub] + o1];
        acc[sub] = wmma4(a, b, acc[sub]);
      }
    }
  }
  if (K4 < K) {  // K tail (only conv1, K=147): fully guarded single step
    int ka = K4 + khalf;
    bool v0 = ka < K, v1 = (ka + 1) < K;
    v2f a;
    a.x = v0 ? Arow0[ka] : 0.f;
    a.y = v1 ? Arow0[ka + 1] : 0.f;
    int o0 = 0, ff0 = 0, o1 = 0, ff1 = 0;
    if (v0) { o0 = sOff[ka]; ff0 = sFF[ka]; }
    if (v1) { o1 = sOff[ka + 1]; ff1 = sFF[ka + 1]; }
    int fy0 = ff0 >> 16, fx0 = ff0 & 0xffff;
    int fy1 = ff1 >> 16, fx1 = ff1 & 0xffff;
#pragma unroll
    for (int sub = 0; sub < 4; ++sub) {
      v2f b;
      b.x = 0.f;
      b.y = 0.f;
      int hi0 = h0[sub] + fy0, wi0 = w0[sub] + fx0;
      if (v0 && (unsigned)hi0 < (unsigned)Hin && (unsigned)wi0 < (unsigned)Win)
        b.x = X[base[sub] + o0];
      int hi1 = h0[sub] + fy1, wi1 = w0[sub] + fx1;
      if (v1 && (unsigned)hi1 < (unsigned)Hin && (unsigned)wi1 < (unsigned)Win)
        b.y = X[base[sub] + o1];
      acc[sub] = wmma4(a, b, acc[sub]);
    }
  }

  const int coHalf = (lane >> 4) * 8;
#pragma unroll
  for (int sub = 0; sub < 4; ++sub) {
    if (!valid[sub]) continue;
    long ob = outBase[sub];
#pragma unroll
    for (int v = 0; v < 8; ++v) {
      int co = coBase + coHalf + v;
      Y[ob + (long)co * HWo] = acc[sub][v];
    }
  }
}

// ---------------------------------------------------------------------------
// Generic strided WMMA GEMM: C[M,N] = alpha * A * B + bias (row-major C).
// A element (m,k) at A[m*sAr + k*sAc]; B element (k,n) at B[k*sBr + n*sBc].
// Full-tile fast path; guarded path only for edge tiles (e.g. N=5994).
// ---------------------------------------------------------------------------
__global__ __launch_bounds__(32) void gemm_wmma(
    const float* __restrict__ A, int sAr, int sAc, const float* __restrict__ Bm,
    int sBr, int sBc, const float* __restrict__ bias, float* __restrict__ C,
    int M, int N, int K, float alpha) {
  const int lane = threadIdx.x;
  const int tM = blockIdx.y * 16, tN = blockIdx.x * 16;
  const int mr = tM + (lane & 15);
  const int nc = tN + (lane & 15);
  const int kh = (lane >> 4) << 1;
  const int K4 = K & ~3;
  const bool full = (tM + 16 <= M) && (tN + 16 <= N);
  v8f acc = {};
  if (full) {
    for (int k0 = 0; k0 < K4; k0 += 4) {
      int ka = k0 + kh;
      v2f a, b;
      a.x = A[(long)mr * sAr + (long)ka * sAc];
      a.y = A[(long)mr * sAr + (long)(ka + 1) * sAc];
      b.x = Bm[(long)ka * sBr + (long)nc * sBc];
      b.y = Bm[(long)(ka + 1) * sBr + (long)nc * sBc];
      acc = wmma4(a, b, acc);
    }
  } else {
    for (int k0 = 0; k0 < K4; k0 += 4) {
      int ka = k0 + kh;
      v2f a, b;
      a.x = a.y = 0.f;
      b.x = b.y = 0.f;
      if (mr < M) {
        a.x = A[(long)mr * sAr + (long)ka * sAc];
        a.y = A[(long)mr * sAr + (long)(ka + 1) * sAc];
      }
      if (nc < N) {
        b.x = Bm[(long)ka * sBr + (long)nc * sBc];
        b.y = Bm[(long)(ka + 1) * sBr + (long)nc * sBc];
      }
      acc = wmma4(a, b, acc);
    }
  }
  if (K4 < K) {  // K tail, fully guarded (not taken by any current call)
    int ka = K4 + kh;
    v2f a, b;
    a.x = a.y = 0.f;
    b.x = b.y = 0.f;
    if (mr < M) {
      if (ka < K) a.x = A[(long)mr * sAr + (long)ka * sAc];
      if (ka + 1 < K) a.y = A[(long)mr * sAr + (long)(ka + 1) * sAc];
    }
    if (nc < N) {
      if (ka < K) b.x = Bm[(long)ka * sBr + (long)nc * sBc];
      if (ka + 1 < K) b.y = Bm[(long)(ka + 1) * sBr + (long)nc * sBc];
    }
    acc = wmma4(a, b, acc);
  }
  float bz = (bias != nullptr && nc < N) ? bias[nc] : 0.f;
  const int rh = tM + (lane >> 4) * 8;
#pragma unroll
  for (int v = 0; v < 8; ++v) {
    int r = rh + v;
    if (r < M && nc < N) C[(long)r * N + nc] = acc[v] * alpha + bz;
  }
}

// ---------------------------------------------------------------------------
// Training-mode BN over (N,H,W): one block per channel -> scale/shift.
// ---------------------------------------------------------------------------
__global__ __launch_bounds__(256) void bn4_stats(
    const float* __restrict__ X, int N, int C, int HW,
    const float* __restrict__ g, const float* __restrict__ b,
    float* __restrict__ sc, float* __restrict__ sh) {
  __shared__ float l1[256], l2[256];
  const int c = blockIdx.x;
  float s = 0.f, s2 = 0.f;
  for (int n = 0; n < N; ++n) {
    const float* p = X + ((long)n * C + c) * (long)HW;
    for (int i = threadIdx.x; i < HW; i += 256) {
      float v = p[i];
      s += v;
      s2 += v * v;
    }
  }
  l1[threadIdx.x] = s;
  l2[threadIdx.x] = s2;
  __syncthreads();
  for (int o = 128; o > 0; o >>= 1) {
    if (threadIdx.x < o) {
      l1[threadIdx.x] += l1[threadIdx.x + o];
      l2[threadIdx.x] += l2[threadIdx.x + o];
    }
    __syncthreads();
  }
  if (threadIdx.x == 0) {
    float cnt = (float)N * (float)HW;
    float m = l1[0] / cnt;
    float var = l2[0] / cnt - m * m;
    float scale = g[c] * rsqrtf(var + EPS_BN);
    sc[c] = scale;
    sh[c] = b[c] - m * scale;
  }
}

// mode: 0 = bn only, 1 = bn+relu, 2 = bn + residual + relu
__global__ __launch_bounds__(256) void bn_apply(
    const float* __restrict__ X, const float* __restrict__ sc,
    const float* __restrict__ sh, const float* __restrict__ res,
    float* __restrict__ out, int C, int HW, int mode) {
  const int nc = blockIdx.y;
  const int c = nc % C;
  const long base = (long)nc * HW;
  const int i = blockIdx.x * 256 + threadIdx.x;
  if (i < HW) {
    float v = X[base + i] * sc[c] + sh[c];
    if (mode == 2) v += res[base + i];
    if (mode >= 1) v = fmaxf(v, 0.f);
    out[base + i] = v;
  }
}

__global__ __launch_bounds__(256) void pool_mean(const float* __restrict__ X,
                                                 float* __restrict__ P,
                                                 int HW) {
  __shared__ float l1[256];
  const long base = (long)blockIdx.x * HW;
  float s = 0.f;
  for (int i = threadIdx.x; i < HW; i += 256) s += X[base + i];
  l1[threadIdx.x] = s;
  __syncthreads();
  for (int o = 128; o > 0; o >>= 1) {
    if (threadIdx.x < o) l1[threadIdx.x] += l1[threadIdx.x + o];
    __syncthreads();
  }
  if (threadIdx.x == 0) P[blockIdx.x] = l1[0] / (float)HW;
}

// qA[c*B + b] = pPrev[b*Cp + c] * mask[c*B + b]    (mask applied to p_prev^T)
__global__ void mask_a_mul(const float* __restrict__ p,
                           const float* __restrict__ mask,
                           float* __restrict__ q, int Cp, int Bn) {
  int i = blockIdx.x * 256 + threadIdx.x;
  if (i < Cp * Bn) {
    int c = i / Bn, b = i - c * Bn;
    q[i] = p[b * Cp + c] * mask[i];
  }
}

__global__ void mask_b_mul(const float* __restrict__ p,
                           const float* __restrict__ mask,
                           float* __restrict__ q, int n) {
  int i = blockIdx.x * 256 + threadIdx.x;
  if (i < n) q[i] = p[i] * mask[i];
}

__global__ __launch_bounds__(256) void softmax_rows(float* __restrict__ X,
                                                    int C, float scale) {
  __shared__ float red[256];
  const int tid = threadIdx.x;
  float* row = X + (long)blockIdx.x * C;
  float mx = -3.4e38f;
  for (int i = tid; i < C; i += 256) mx = fmaxf(mx, row[i] * scale);
  red[tid] = mx;
  __syncthreads();
  for (int o = 128; o > 0; o >>= 1) {
    if (tid < o) red[tid] = fmaxf(red[tid], red[tid + o]);
    __syncthreads();
  }
  float m = red[0];
  __syncthreads();
  float s = 0.f;
  for (int i = tid; i < C; i += 256) s += __expf(row[i] * scale - m);
  red[tid] = s;
  __syncthreads();
  for (int o = 128; o > 0; o >>= 1) {
    if (tid < o) red[tid] += red[tid + o];
    __syncthreads();
  }
  float inv = 1.f / red[0];
  for (int i = tid; i < C; i += 256) row[i] = __expf(row[i] * scale - m) * inv;
}

// BN over batch dim (R rows) + leaky relu; one thread per column.
__global__ void bn1_lrelu(const float* __restrict__ in, float* __restrict__ out,
                          const float* __restrict__ g,
                          const float* __restrict__ b, int R, int C) {
  int c = blockIdx.x * blockDim.x + threadIdx.x;
  if (c >= C) return;
  float s = 0.f, s2 = 0.f;
  for (int r = 0; r < R; ++r) {
    float v = in[(long)r * C + c];
    s += v;
    s2 += v * v;
  }
  float m = s / R;
  float var = s2 / R - m * m;
  float sc = g[c] * rsqrtf(var + EPS_BN);
  float sh = b[c] - m * sc;
  for (int r = 0; r < R; ++r) {
    float v = in[(long)r * C + c] * sc + sh;
    out[(long)r * C + c] = v > 0.f ? v : 0.01f * v;
  }
}

__global__ void concat2(const float* __restrict__ a, const float* __restrict__ b,
                        float* __restrict__ out, int R, int Ca, int Cb) {
  int i = blockIdx.x * 256 + threadIdx.x;
  int Ct = Ca + Cb;
  if (i < R * Ct) {
    int r = i / Ct, c = i - r * Ct;
    out[i] = (c < Ca) ? a[r * Ca + c] : b[r * Cb + (c - Ca)];
  }
}

__global__ __launch_bounds__(256) void log_softmax_rows(
    const float* __restrict__ in, float* __restrict__ out, int C) {
  __shared__ float red[256];
  const int tid = threadIdx.x;
  const float* row = in + (long)blockIdx.x * C;
  float* orow = out + (long)blockIdx.x * C;
  float mx = -3.4e38f;
  for (int i = tid; i < C; i += 256) mx = fmaxf(mx, row[i]);
  red[tid] = mx;
  __syncthreads();
  for (int o = 128; o > 0; o >>= 1) {
    if (tid < o) red[tid] = fmaxf(red[tid], red[tid + o]);
    __syncthreads();
  }
  float m = red[0];
  __syncthreads();
  float s = 0.f;
  for (int i = tid; i < C; i += 256) s += __expf(row[i] - m);
  red[tid] = s;
  __syncthreads();
  for (int o = 128; o > 0; o >>= 1) {
    if (tid < o) red[tid] += red[tid + o];
    __syncthreads();
  }
  float lg = logf(red[0]);
  for (int i = tid; i < C; i += 256) orow[i] = row[i] - m - lg;
}

// ---------------------------------------------------------------------------
extern "C" void kernel_launch(void* const* d_in, const int* in_sizes, int n_in,
                              void* d_out, int out_size, void* d_ws,
                              size_t ws_size, hipStream_t stream) {
  (void)in_sizes; (void)n_in; (void)out_size; (void)ws_size;
  auto F = [&](int i) { return (const float*)d_in[i]; };
  float* ws = (float*)d_ws;
  const long BIG = 25690112L;  // 32*64*112*112
  float* P = ws;
  float* Q = ws + BIG;
  float* R = ws + 2 * BIG;
  long off = 3 * BIG;
  auto alloc = [&](long n) {
    float* p = ws + off;
    off += (n + 63) & ~63L;
    return p;
  };
  float* sc1 = alloc(512); float* sh1 = alloc(512);
  float* sc2 = alloc(512); float* sh2 = alloc(512);
  float* sc3 = alloc(512); float* sh3 = alloc(512);
  float* pv[5] = {alloc(32 * 64), alloc(32 * 64), alloc(32 * 128),
                  alloc(32 * 256), alloc(32 * 512)};
  float* qA = alloc(256 * 32);
  float* qB = alloc(32 * 256);
  float* S1 = alloc(256 * 512);
  float* S2 = alloc(512 * 256);
  float* qm = alloc(256 * 32);
  float* qmc = alloc(512 * 32);
  float* zsb[4] = {alloc(64 * 64), alloc(64 * 128), alloc(128 * 256),
                   alloc(256 * 512)};
  float* za = alloc(32 * 512);
  float* zb = alloc(32 * 512);
  float* cat = alloc(32 * 1024);
  float* t1 = alloc(32 * 512);
  float* t2 = alloc(32 * 512);
  float* t3 = alloc(32 * 512);
  float* logits = alloc(32L * 5994);

  auto conv = [&](const float* X, const float* W, float* Y, int N, int Cin,
                  int Hin, int Win, int Cout, int KH, int KW, int st, int pad,
                  int Ho, int Wo) {
    long Ngemm = (long)N * Ho * Wo;
    dim3 g((unsigned)((Ngemm + 511) / 512), (unsigned)(Cout / 16));
    conv_wmma<<<g, 256, 0, stream>>>(X, W, Y, N, Cin, Hin, Win, Cout, KH, KW,
                                     st, pad, Ho, Wo);
  };
  auto bnstats = [&](const float* X, int C, int HW, const float* g,
                     const float* b, float* sc, float* sh) {
    bn4_stats<<<C, 256, 0, stream>>>(X, 32, C, HW, g, b, sc, sh);
  };
  auto bnapply = [&](const float* X, const float* sc, const float* sh,
                     const float* res, float* out, int NC, int C, int HW,
                     int mode) {
    dim3 g((unsigned)((HW + 255) / 256), (unsigned)NC);
    bn_apply<<<g, 256, 0, stream>>>(X, sc, sh, res, out, C, HW, mode);
  };
  auto gemm = [&](const float* A, int sAr, int sAc, const float* Bm, int sBr,
                  int sBc, const float* bias, float* C, int M, int N, int K,
                  float alpha) {
    dim3 g((unsigned)((N + 15) / 16), (unsigned)((M + 15) / 16));
    gemm_wmma<<<g, 32, 0, stream>>>(A, sAr, sAc, Bm, sBr, sBc, bias, C, M, N,
                                    K, alpha);
  };
  auto cross = [&](const float* pPrev, int Cp, const float* pCur, int Cc,
                   const float* mA, const float* mB, float* zout) {
    int nA = Cp * 32;
    mask_a_mul<<<(nA + 255) / 256, 256, 0, stream>>>(pPrev, mA, qA, Cp, 32);
    mask_b_mul<<<(nA + 255) / 256, 256, 0, stream>>>(pPrev, mB, qB, nA);
    // S1 = (p_prev^T * mask_a) @ p_cur     [Cp,Cc], softmax/sqrt(Cc)
    gemm(qA, 32, 1, pCur, Cc, 1, nullptr, S1, Cp, Cc, 32, 1.f);
    softmax_rows<<<Cp, 256, 0, stream>>>(S1, Cc, 1.f / sqrtf((float)Cc));
    // q_m = S1 @ p_cur^T                   [Cp,32]
    gemm(S1, Cc, 1, pCur, 1, Cc, nullptr, qm, Cp, 32, Cc, 1.f);
    // S2 = p_cur^T @ (p_prev * mask_b)     [Cc,Cp], softmax/sqrt(Cp)
    gemm(pCur, 1, Cc, qB, Cp, 1, nullptr, S2, Cc, Cp, 32, 1.f);
    softmax_rows<<<Cc, 256, 0, stream>>>(S2, Cp, 1.f / sqrtf((float)Cp));
    // q_m_c = S2 @ p_prev^T                [Cc,32]
    gemm(S2, Cp, 1, pPrev, 1, Cp, nullptr, qmc, Cc, 32, Cp, 1.f);
    // z = q_m @ q_m_c^T                    [Cp,Cc]
    gemm(qm, 32, 1, qmc, 1, 32, nullptr, zout, Cp, Cc, 32, 1.f);
  };

  struct BlkIdx { int w1, g1, b1, w2, g2, b2, wd, gd, bd; };
  const BlkIdx blks[4][2] = {
      {{4, 5, 6, 7, 8, 9, -1, -1, -1}, {10, 11, 12, 13, 14, 15, -1, -1, -1}},
      {{16, 17, 18, 19, 20, 21, 22, 23, 24},
       {25, 26, 27, 28, 29, 30, -1, -1, -1}},
      {{31, 32, 33, 34, 35, 36, 37, 38, 39},
       {40, 41, 42, 43, 44, 45, -1, -1, -1}},
      {{46, 47, 48, 49, 50, 51, 52, 53, 54},
       {55, 56, 57, 58, 59, 60, -1, -1, -1}}};
  const int chin[4] = {64, 64, 128, 256};
  const int chout[4] = {64, 128, 256, 512};
  const int strd[4] = {1, 2, 2, 2};

  // ---- stem: conv1 7x7/2 + BN + ReLU + pool -> p0
  conv(F(0), F(1), P, 32, 3, 224, 224, 64, 7, 7, 2, 3, 112, 112);
  bnstats(P, 64, 112 * 112, F(2), F(3), sc1, sh1);
  bnapply(P, sc1, sh1, nullptr, P, 32 * 64, 64, 112 * 112, 1);
  pool_mean<<<32 * 64, 256, 0, stream>>>(P, pv[0], 112 * 112);

  // ---- residual stages
  int curH = 112;
  for (int s = 0; s < 4; ++s) {
    for (int j = 0; j < 2; ++j) {
      const BlkIdx& bl = blks[s][j];
      int ic = j ? chout[s] : chin[s];
      int oc = chout[s];
      int st = j ? 1 : strd[s];
      int hi = curH, ho = hi / st;
      conv(P, F(bl.w1), Q, 32, ic, hi, hi, oc, 3, 3, st, 1, ho, ho);
      bnstats(Q, oc, ho * ho, F(bl.g1), F(bl.b1), sc1, sh1);
      bnapply(Q, sc1, sh1, nullptr, Q, 32 * oc, oc, ho * ho, 1);
      conv(Q, F(bl.w2), R, 32, oc, ho, ho, oc, 3, 3, 1, 1, ho, ho);
      bnstats(R, oc, ho * ho, F(bl.g2), F(bl.b2), sc2, sh2);
      const float* res = P;
      if (bl.wd >= 0) {
        conv(P, F(bl.wd), Q, 32, ic, hi, hi, oc, 1, 1, st, 0, ho, ho);
        bnstats(Q, oc, ho * ho, F(bl.gd), F(bl.bd), sc3, sh3);
        bnapply(Q, sc3, sh3, nullptr, Q, 32 * oc, oc, ho * ho, 0);
        res = Q;
      }
      bnapply(R, sc2, sh2, res, P, 32 * oc, oc, ho * ho, 2);
      curH = ho;
    }
    pool_mean<<<32 * chout[s], 256, 0, stream>>>(P, pv[s + 1], curH * curH);
    cross(pv[s], chin[s], pv[s + 1], chout[s], F(77 + 2 * s), F(78 + 2 * s),
          zsb[s]);
  }

  // ---- z chain: p0 @ zs0 @ zs1 @ zs2 @ zs3 -> [32,512]
  const float* zcur = pv[0];
  int zw = 64;
  float* zo = za;
  for (int s = 0; s < 4; ++s) {
    int cc = chout[s];
    gemm(zcur, zw, 1, zsb[s], cc, 1, nullptr, zo, 32, cc, zw, 1.f);
    zcur = zo;
    zw = cc;
    zo = (zo == za) ? zb : za;
  }

  // ---- head
  concat2<<<(32 * 1024 + 255) / 256, 256, 0, stream>>>(zcur, pv[4], cat, 32,
                                                       512, 512);
  bn1_lrelu<<<4, 256, 0, stream>>>(cat, cat, F(61), F(62), 32, 1024);
  gemm(cat, 1024, 1, F(63), 1, 1024, F(64), t1, 32, 512, 1024, 1.f);
  bn1_lrelu<<<2, 256, 0, stream>>>(t1, t1, F(65), F(66), 32, 512);
  gemm(t1, 512, 1, F(67), 1, 512, F(68), t2, 32, 512, 512, 1.f);
  bn1_lrelu<<<2, 256, 0, stream>>>(t2, t2, F(69), F(70), 32, 512);
  float* spk = (float*)d_out;  // output #1: fc3 pre-BN activations
  gemm(t2, 512, 1, F(71), 1, 512, F(72), spk, 32, 512, 512, 1.f);
  bn1_lrelu<<<2, 256, 0, stream>>>(spk, t3, F(73), F(74), 32, 512);
  gemm(t3, 512, 1, F(75), 1, 512, F(76), logits, 32, 5994, 512, 1.f);
  log_softmax_rows<<<32, 256, 0, stream>>>(logits, (float*)d_out + 32 * 512,
                                           5994);
}